// BaseExecutionBackend_9036611190826
// MI455X (gfx1250) — compile-verified
//
#include <hip/hip_runtime.h>
#include <hip/hip_bf16.h>
#include <stdint.h>

typedef __attribute__((ext_vector_type(2))) float v2f;
typedef __attribute__((ext_vector_type(8))) float v8f;

#define L_COEFF 9
#define C_IN 64
#define C_MSG 128
#define X_ROW_STRIDE (L_COEFF * C_IN)      // 576 floats per node
#define OUT_EDGE_STRIDE (L_COEFF * C_MSG)  // 1152 floats per edge

// Fast sigmoid: v_exp_f32 + v_add + v_rcp_f32 (no IEEE divide chain).
__device__ __forceinline__ float sigmoidf_fast(float x) {
    return __builtin_amdgcn_rcpf(1.0f + __expf(-x));
}

// One block (128 threads = 4 wave32) per edge.
// Stage 1: combined = W_inv @ W (9x9) in LDS, zero-padded to 16x12 (A-matrix staging).
//          (matmul associativity: W_inv @ (W @ X) == (W_inv @ W) @ X — halves the
//          big-matmul FLOPs and removes the chained-WMMA layout shuffle)
// Stage 2: back = combined @ X via v_wmma_f32_16x16x4_f32, 8 N-tiles of 16 channels,
//          K = 9 padded to 12 -> 3 WMMAs per tile, 2 tiles per wave.
// Stage 3: fused epilogue (silu on row 0, sigmoid gating on rows 1..8) straight
//          from the WMMA C/D register layout.
__global__ __launch_bounds__(128)
void eqv2_edge_wmma_kernel(const float* __restrict__ x_full,
                           const float* __restrict__ wigner,
                           const float* __restrict__ wigner_inv,
                           const float* __restrict__ gating,
                           const long long* __restrict__ edge_index,
                           float* __restrict__ out,
                           int n_edges)
{
    const int e = blockIdx.x;
    if (e >= n_edges) return;

    __shared__ float sW[81];
    __shared__ float sWinv[81];
    __shared__ float sC[16 * 12];   // combined matrix, zero padded, row-major [M][K]

    const int tid = threadIdx.x;

    // Zero the padded staging buffer, cooperatively load this edge's Wigner pair.
    for (int i = tid; i < 16 * 12; i += 128) sC[i] = 0.0f;
    const float* wp  = wigner     + (size_t)e * 81;
    const float* wip = wigner_inv + (size_t)e * 81;
    for (int i = tid; i < 81; i += 128) { sW[i] = wp[i]; sWinv[i] = wip[i]; }
    __syncthreads();

    // combined[l][k] = sum_m Winv[l][m] * W[m][k]   (tiny: 81 dots of length 9)
    if (tid < 81) {
        const int l = tid / 9, k = tid % 9;
        float acc = 0.0f;
        #pragma unroll
        for (int m = 0; m < 9; ++m)
            acc = fmaf(sWinv[l * 9 + m], sW[m * 9 + k], acc);
        sC[l * 12 + k] = acc;
    }
    __syncthreads();

    const long long src = edge_index[e];
    const long long dst = edge_index[(size_t)n_edges + e];
    const float* xs = x_full + (size_t)src * X_ROW_STRIDE;
    const float* xt = x_full + (size_t)dst * X_ROW_STRIDE;

    const int lane = tid & 31;
    const int wid  = tid >> 5;
    const int half = lane >> 4;     // A/B layout: lane half selects K pair
    const int l16  = lane & 15;

    // A-fragments: 16x4 f32 tiles of the combined matrix.
    // Lane layout (ISA 16x4 f32 A): M = lane&15, VGPR j holds K = 4*kb + 2*half + j.
    v2f afrag[3];
    #pragma unroll
    for (int kb = 0; kb < 3; ++kb) {
        const int k0 = 4 * kb + 2 * half;
        afrag[kb].x = sC[l16 * 12 + k0];
        afrag[kb].y = sC[l16 * 12 + k0 + 1];
    }

    const float* gptr  = gating + (size_t)e * 256;
    float*       obase = out    + (size_t)e * OUT_EDGE_STRIDE;

    #pragma unroll
    for (int t = 0; t < 2; ++t) {
        const int nbase = (wid * 2 + t) * 16;
        const int n = nbase + l16;
        // Column n of the concatenated message matrix X (9 x 128):
        // first 64 channels from source node, last 64 from target node.
        const float* xcol = (n < C_IN) ? (xs + n) : (xt + (n - C_IN));

        v8f acc = {};
        #pragma unroll
        for (int kb = 0; kb < 3; ++kb) {
            const int kA = 4 * kb + 2 * half;      // B layout mirrors A: K in (half, vgpr)
            v2f bfrag;
            bfrag.x = (kA     < L_COEFF) ? xcol[(size_t)kA * C_IN]       : 0.0f;
            bfrag.y = (kA + 1 < L_COEFF) ? xcol[(size_t)(kA + 1) * C_IN] : 0.0f;
            acc = __builtin_amdgcn_wmma_f32_16x16x4_f32(
                /*neg_a=*/false, afrag[kb], /*neg_b=*/false, bfrag,
                /*c_mod=*/(short)0, acc, /*reuse_a=*/false, /*reuse_b=*/false);
        }

        // gate rows: g0/g1 = sigmoid of the two 128-channel gating rows
        const float g0 = sigmoidf_fast(gptr[n]);
        const float g1 = sigmoidf_fast(gptr[128 + n]);

        // C/D layout: lanes 0-15 hold rows 0..7 (VGPR v = row v),
        //             lanes 16-31 hold rows 8..15 (VGPR v = row 8+v).
        if (half == 0) {
            const float r0 = acc[0];
            obase[0 * C_MSG + n] = r0 * sigmoidf_fast(r0);   // silu(row 0)
            obase[1 * C_MSG + n] = acc[1] * g0;              // EXPAND_INDEX[0]=0
            obase[2 * C_MSG + n] = acc[2] * g1;              // EXPAND_INDEX[1]=1
            obase[3 * C_MSG + n] = acc[3] * g0;              // EXPAND_INDEX[2]=0
            obase[4 * C_MSG + n] = acc[4] * g1;              // EXPAND_INDEX[3]=1
            obase[5 * C_MSG + n] = acc[5] * g0;              // EXPAND_INDEX[4]=0
            obase[6 * C_MSG + n] = acc[6] * g1;              // EXPAND_INDEX[5]=1
            obase[7 * C_MSG + n] = acc[7] * g1;              // EXPAND_INDEX[6]=1
        } else {
            obase[8 * C_MSG + n] = acc[0] * g1;              // EXPAND_INDEX[7]=1
        }
    }
}

extern "C" void kernel_launch(void* const* d_in, const int* in_sizes, int n_in,
                              void* d_out, int out_size, void* d_ws, size_t ws_size,
                              hipStream_t stream)
{
    const float*     x_full = (const float*)d_in[0];
    const float*     wigner = (const float*)d_in[1];
    const float*     winv   = (const float*)d_in[2];
    const float*     gating = (const float*)d_in[3];
    const long long* eidx   = (const long long*)d_in[4];
    float*           outp   = (float*)d_out;

    const int n_edges = in_sizes[4] / 2;   // edge_index is (2, E)

    dim3 grid((unsigned)n_edges), block(128);
    hipLaunchKernelGGL(eqv2_edge_wmma_kernel, grid, block, 0, stream,
                       x_full, wigner, winv, gating, eidx, outp, n_edges);
}